// GNN_60120952209896
// MI455X (gfx1250) — compile-verified
//
#include <hip/hip_runtime.h>
#include <math.h>

#define N_SURF 40000
#define N_LIG  1024
#define HIDD   512
#define TDIMM  64
#define EMAX   131072

typedef float v2f __attribute__((ext_vector_type(2)));
typedef float v8f __attribute__((ext_vector_type(8)));

// ---------------- init: deg = 1, agg = 0 ----------------
__global__ void k_init(float* deg, float* agg) {
    int i = blockIdx.x * blockDim.x + threadIdx.x;   // 4096 threads
    if (i < N_LIG) deg[i] = 1.0f;
    if (i < N_LIG * 4) agg[i] = 0.0f;
}

// ---------------- degree accumulation ----------------
__global__ void k_deg(const int* eidx, float* deg) {
    int e = blockIdx.x * blockDim.x + threadIdx.x;
    if (e >= EMAX) return;
    int src = eidx[e];
    int dst = eidx[EMAX + e];
    // w = (~mask_ligand[src] & mask_ligand[dst])
    if (src < N_SURF && dst >= N_SURF && dst < N_SURF + N_LIG)
        atomicAdd(deg + (dst - N_SURF), 1.0f);
}

// ---------------- dinv ----------------
__global__ void k_dinv(const float* deg, float* dinv) {
    int i = blockIdx.x * blockDim.x + threadIdx.x;
    if (i < N_LIG) dinv[i] = rsqrtf(deg[i]);
}

// ---------------- edge aggregation: agg[l] = {sum nrm*pos, sum nrm} ----------------
// src is a surface node -> dinv[src] == 1, so norm = dinv[dst]
__global__ void k_agg(const int* eidx, const float* surf_pos, const float* dinv, float* agg) {
    int e = blockIdx.x * blockDim.x + threadIdx.x;
    if (e >= EMAX) return;
    int src = eidx[e];
    int dst = eidx[EMAX + e];
    if (src < N_SURF && dst >= N_SURF && dst < N_SURF + N_LIG) {
        int l = dst - N_SURF;
        float nrm = dinv[l];
        atomicAdd(agg + l * 4 + 0, nrm * surf_pos[src * 3 + 0]);
        atomicAdd(agg + l * 4 + 1, nrm * surf_pos[src * 3 + 1]);
        atomicAdd(agg + l * 4 + 2, nrm * surf_pos[src * 3 + 2]);
        atomicAdd(agg + l * 4 + 3, nrm);
    }
}

// ---------------- fuse W_gcn[5] @ W_pos (512x3, padded to 512x16) + bias ----------------
__global__ void k_fuse(const float* W_gcn, const float* b_gcn,
                       const float* W_pos, const float* b_pos,
                       float* wf, float* bf) {
    int idx = blockIdx.x * blockDim.x + threadIdx.x;
    const float* Wg = W_gcn + 5 * HIDD * HIDD;
    if (idx < HIDD * 16) {
        int k = idx >> 4, c = idx & 15;
        float s = 0.0f;
        if (c < 3)
            for (int j = 0; j < HIDD; ++j)
                s = fmaf(Wg[k * HIDD + j], W_pos[j * 3 + c], s);
        wf[idx] = s;
    } else if (idx < HIDD * 16 + 3) {
        int c = idx - HIDD * 16;
        float s = b_pos[c];
        const float* bg = b_gcn + 5 * HIDD;
        for (int j = 0; j < HIDD; ++j)
            s = fmaf(bg[j], W_pos[j * 3 + c], s);
        bf[c] = s;
    }
}

// ---------------- per-ligand features -> H_eff = agg_h + dinv^2 * h_lig ----------------
__global__ void k_feat(const float* lig_pos, const float* time_,
                       const float* W_t1, const float* b_t1,
                       const float* W_t2, const float* b_t2,
                       const float* W_lig, const float* b_lig,
                       const float* W_gate, const float* b_gate,
                       const float* W_hbias,
                       const float* W_surf, const float* b_surf,
                       const float* agg, const float* dinv,
                       float* heff) {
    __shared__ float emb[TDIMM];
    __shared__ float hid[4 * TDIMM];
    __shared__ float ht[TDIMM];
    int i = blockIdx.x;          // ligand node
    int tid = threadIdx.x;       // 256 threads
    float t = time_[i];
    if (tid < 32) {
        float f = expf(-logf(10000.0f) / 31.0f * (float)tid);
        float ang = t * f;
        emb[tid]      = sinf(ang);
        emb[tid + 32] = cosf(ang);
    }
    __syncthreads();
    {   // hidden = gelu(emb @ W_t1 + b_t1), 256 wide
        float s = b_t1[tid];
        for (int k = 0; k < TDIMM; ++k) s = fmaf(emb[k], W_t1[k * 256 + tid], s);
        hid[tid] = 0.5f * s * (1.0f + erff(s * 0.70710678118654752f));
    }
    __syncthreads();
    if (tid < TDIMM) {   // h_time = hidden @ W_t2 + b_t2
        float s = b_t2[tid];
        for (int k = 0; k < 256; ++k) s = fmaf(hid[k], W_t2[k * TDIMM + tid], s);
        ht[tid] = s;
    }
    __syncthreads();
    float p0 = lig_pos[i * 3 + 0], p1 = lig_pos[i * 3 + 1], p2 = lig_pos[i * 3 + 2];
    float a0 = agg[i * 4 + 0], a1 = agg[i * 4 + 1], a2 = agg[i * 4 + 2], wsum = agg[i * 4 + 3];
    float di = dinv[i], d2 = di * di;
    for (int jj = tid; jj < HIDD; jj += 256) {
        float g = b_gate[jj], hb = 0.0f;
        for (int k = 0; k < TDIMM; ++k) {
            float h = ht[k];
            g  = fmaf(h, W_gate[k * HIDD + jj], g);
            hb = fmaf(h, W_hbias[k * HIDD + jj], hb);
        }
        float hl = b_lig[jj] + p0 * W_lig[jj] + p1 * W_lig[HIDD + jj] + p2 * W_lig[2 * HIDD + jj];
        float sg = 1.0f / (1.0f + expf(-g));
        hl = hl * sg + hb;
        float ah = wsum * b_surf[jj] + a0 * W_surf[jj] + a1 * W_surf[HIDD + jj] + a2 * W_surf[2 * HIDD + jj];
        heff[i * HIDD + jj] = ah + d2 * hl;
    }
}

// ---------------- out = H_eff(1024x512) @ Wf(512x16) + bf, via V_WMMA_F32_16X16X4_F32 ----------------
// One wave (32 lanes) per 16-row tile; K loop of 128 WMMA ops.
// A (16x4 f32): lanes 0-15 hold M=lane, K = k0+{0,1}; lanes 16-31 hold M=lane-16, K = k0+{2,3}.
// B (4x16 f32): mirrored — VGPR v, lane-half h: K = k0 + 2h + v, N = lane&15.
// C/D: VGPR r holds row r (lanes 0-15) / r+8 (lanes 16-31), col = lane&15.
__global__ void k_out(const float* __restrict__ heff, const float* __restrict__ wf,
                      const float* __restrict__ bf, float* __restrict__ out) {
    int tile = blockIdx.x;          // 64 tiles of 16 rows
    int lane = threadIdx.x;         // 32 lanes
    int half = lane >> 4;
    int m    = lane & 15;
    int n    = lane & 15;
    v8f c = {};
    const float* arow = heff + (tile * 16 + m) * HIDD;
    for (int k0 = 0; k0 < HIDD; k0 += 4) {
        v2f a, b;
        int ka = k0 + 2 * half;
        a[0] = arow[ka + 0];
        a[1] = arow[ka + 1];
        b[0] = wf[(ka + 0) * 16 + n];
        b[1] = wf[(ka + 1) * 16 + n];
        c = __builtin_amdgcn_wmma_f32_16x16x4_f32(false, a, false, b,
                                                  (short)0, c, false, false);
    }
    float bias = (n < 3) ? bf[n] : 0.0f;
    for (int r = 0; r < 8; ++r) {
        int row = tile * 16 + r + 8 * half;
        if (n < 3) out[row * 3 + n] = c[r] + bias;
    }
}

extern "C" void kernel_launch(void* const* d_in, const int* in_sizes, int n_in,
                              void* d_out, int out_size, void* d_ws, size_t ws_size,
                              hipStream_t stream) {
    const float* surf_pos = (const float*)d_in[0];
    const float* lig_pos  = (const float*)d_in[1];
    const float* time_    = (const float*)d_in[2];
    const int*   eidx     = (const int*)  d_in[3];
    const float* W_surf   = (const float*)d_in[6];
    const float* b_surf   = (const float*)d_in[7];
    const float* W_t1     = (const float*)d_in[8];
    const float* b_t1     = (const float*)d_in[9];
    const float* W_t2     = (const float*)d_in[10];
    const float* b_t2     = (const float*)d_in[11];
    const float* W_lig    = (const float*)d_in[12];
    const float* b_lig    = (const float*)d_in[13];
    const float* W_gate   = (const float*)d_in[14];
    const float* b_gate   = (const float*)d_in[15];
    const float* W_hbias  = (const float*)d_in[16];
    const float* W_gcn    = (const float*)d_in[17];
    const float* b_gcn    = (const float*)d_in[18];
    const float* W_pos    = (const float*)d_in[19];
    const float* b_pos    = (const float*)d_in[20];

    float* ws   = (float*)d_ws;
    float* deg  = ws;                // 1024
    float* dinv = ws + 1024;         // 1024
    float* agg  = ws + 2048;         // 1024*4
    float* wf   = ws + 8192;         // 512*16
    float* bf   = ws + 16384;        // 3
    float* heff = ws + 16896;        // 1024*512
    float* out  = (float*)d_out;

    k_init<<<16, 256, 0, stream>>>(deg, agg);
    k_deg <<<EMAX / 256, 256, 0, stream>>>(eidx, deg);
    k_dinv<<<4, 256, 0, stream>>>(deg, dinv);
    k_agg <<<EMAX / 256, 256, 0, stream>>>(eidx, surf_pos, dinv, agg);
    k_fuse<<<33, 256, 0, stream>>>(W_gcn, b_gcn, W_pos, b_pos, wf, bf);
    k_feat<<<N_LIG, 256, 0, stream>>>(lig_pos, time_, W_t1, b_t1, W_t2, b_t2,
                                      W_lig, b_lig, W_gate, b_gate, W_hbias,
                                      W_surf, b_surf, agg, dinv, heff);
    k_out <<<N_LIG / 16, 32, 0, stream>>>(heff, wf, bf, out);
}